// GINNet_82197084111148
// MI455X (gfx1250) — compile-verified
//
#include <hip/hip_runtime.h>

#define N_NODES 50000
#define N_EDGES 800000
#define DIM     128
#define CLS     40

typedef __attribute__((ext_vector_type(2))) float v2f;
typedef __attribute__((ext_vector_type(8))) float v8f;

// ---------------------------------------------------------------- zero fill
__global__ void zero_f32(float* __restrict__ p, int n4) {
  int i = blockIdx.x * blockDim.x + threadIdx.x;
  if (i < n4) {
    float4 z = make_float4(0.f, 0.f, 0.f, 0.f);
    *(float4*)(p + (size_t)i * 4) = z;
  }
}

// ------------------------------------------------------ edge scatter (agg)
// One wave32 per edge: 32 lanes x float4 = 128 features.
__global__ void scatter_add(const float* __restrict__ feat,
                            const int* __restrict__ src,
                            const int* __restrict__ dst,
                            float* __restrict__ agg, int nEdges) {
  const int wave = threadIdx.x >> 5;
  const int lane = threadIdx.x & 31;
  const int e = blockIdx.x * (blockDim.x >> 5) + wave;
  if (e >= nEdges) return;
  const int s = src[e];
  const int d = dst[e];
  const float4 v = *(const float4*)(feat + (size_t)s * DIM + lane * 4);
  float* o = agg + (size_t)d * DIM + lane * 4;
  unsafeAtomicAdd(o + 0, v.x);
  unsafeAtomicAdd(o + 1, v.y);
  unsafeAtomicAdd(o + 2, v.z);
  unsafeAtomicAdd(o + 3, v.w);
}

// ------------------------------------------- layer 1: relu((x+agg) @ W1^T)
// grid.x = 3125 row tiles; 8 waves per block = 8 column tiles (128 cols).
__global__ void gin_gemm1(const float* __restrict__ x,
                          const float* __restrict__ agg,
                          const float* __restrict__ W,   // [128,128] (out,in)
                          float* __restrict__ h1) {
  const int wave    = threadIdx.x >> 5;
  const int lane    = threadIdx.x & 31;
  const int rowTile = blockIdx.x;
  const int colTile = wave;
  const int koff    = (lane >> 4) * 2;     // K sub-offset per ISA A/B layout
  const size_t arow = (size_t)(rowTile * 16 + (lane & 15)) * DIM;
  const size_t brow = (size_t)(colTile * 16 + (lane & 15)) * DIM;

  v8f c = {};
#pragma unroll
  for (int kb = 0; kb < DIM; kb += 4) {
    const float2 xa = *(const float2*)(x   + arow + kb + koff);
    const float2 ga = *(const float2*)(agg + arow + kb + koff);
    const float2 wb = *(const float2*)(W   + brow + kb + koff);
    v2f a, b;
    a.x = xa.x + ga.x;  a.y = xa.y + ga.y;   // h = (1+eps)*x + agg, eps=0
    b.x = wb.x;         b.y = wb.y;
    c = __builtin_amdgcn_wmma_f32_16x16x4_f32(false, a, false, b,
                                              (short)0, c, false, false);
  }

  // C/D layout: VGPR v, lanes<16 -> M=v, lanes>=16 -> M=v+8; N = lane&15.
  const int n  = lane & 15;
  const int mh = (lane >> 4) * 8;
#pragma unroll
  for (int v = 0; v < 8; ++v) {
    float val = c[v];
    val = val > 0.f ? val : 0.f;             // fused ReLU
    h1[(size_t)(rowTile * 16 + mh + v) * DIM + colTile * 16 + n] = val;
  }
}

// --------------------------------------------- layer 2: (h1+agg2) @ W2^T
// grid.x = 3125 row tiles; 3 waves per block = 3 column tiles (48 >= 40).
__global__ void gin_gemm2(const float* __restrict__ h1,
                          const float* __restrict__ agg,
                          const float* __restrict__ W2,  // [40,128]
                          float* __restrict__ out) {
  const int wave    = threadIdx.x >> 5;
  const int lane    = threadIdx.x & 31;
  const int rowTile = blockIdx.x;
  const int colTile = wave;
  const int koff    = (lane >> 4) * 2;
  const int nglob   = colTile * 16 + (lane & 15);
  const int nsafe   = (nglob < CLS) ? nglob : 0;          // clamp, no divergence
  const float msk   = (nglob < CLS) ? 1.f : 0.f;          // zero padded B cols
  const size_t arow = (size_t)(rowTile * 16 + (lane & 15)) * DIM;
  const size_t brow = (size_t)nsafe * DIM;

  v8f c = {};
#pragma unroll
  for (int kb = 0; kb < DIM; kb += 4) {
    const float2 xa = *(const float2*)(h1  + arow + kb + koff);
    const float2 ga = *(const float2*)(agg + arow + kb + koff);
    const float2 wb = *(const float2*)(W2  + brow + kb + koff);
    v2f a, b;
    a.x = xa.x + ga.x;  a.y = xa.y + ga.y;
    b.x = wb.x * msk;   b.y = wb.y * msk;
    c = __builtin_amdgcn_wmma_f32_16x16x4_f32(false, a, false, b,
                                              (short)0, c, false, false);
  }

  const int n  = lane & 15;
  const int mh = (lane >> 4) * 8;
  const int col = colTile * 16 + n;
  if (col < CLS) {
#pragma unroll
    for (int v = 0; v < 8; ++v)
      out[(size_t)(rowTile * 16 + mh + v) * CLS + col] = c[v];
  }
}

// --------------------------------------------------------------- launcher
extern "C" void kernel_launch(void* const* d_in, const int* in_sizes, int n_in,
                              void* d_out, int out_size, void* d_ws, size_t ws_size,
                              hipStream_t stream) {
  const float* x    = (const float*)d_in[0];
  const int*   edge = (const int*)d_in[1];     // [2, 800000] flat
  const float* W1   = (const float*)d_in[2];
  const float* W2   = (const float*)d_in[3];
  float*       outp = (float*)d_out;

  const int*   src  = edge;
  const int*   dst  = edge + N_EDGES;

  float* agg = (float*)d_ws;                   // 25.6 MB, reused both layers
  float* h1  = agg + (size_t)N_NODES * DIM;    // 25.6 MB

  const int featN  = N_NODES * DIM;            // 6.4M floats
  const int zBlk   = (featN / 4 + 255) / 256;  // float4 per thread
  const int sBlk   = (N_EDGES + 7) / 8;        // 8 edges (waves) per block

  // ---- layer 1
  zero_f32<<<zBlk, 256, 0, stream>>>(agg, featN / 4);
  scatter_add<<<sBlk, 256, 0, stream>>>(x, src, dst, agg, N_EDGES);
  gin_gemm1<<<N_NODES / 16, 256, 0, stream>>>(x, agg, W1, h1);

  // ---- layer 2
  zero_f32<<<zBlk, 256, 0, stream>>>(agg, featN / 4);
  scatter_add<<<sBlk, 256, 0, stream>>>(h1, src, dst, agg, N_EDGES);
  gin_gemm2<<<N_NODES / 16, 96, 0, stream>>>(h1, agg, W2, outp);
}